// RNNAttention_89498528514716
// MI455X (gfx1250) — compile-verified
//
#include <hip/hip_runtime.h>
#include <hip/hip_bf16.h>

typedef __attribute__((ext_vector_type(2))) float v2f;
typedef __attribute__((ext_vector_type(8))) float v8f;

#define NN 320
#define FB_S 328   // F panel LDS row stride (floats), 16B-aligned rows, conflict-free halves
#define WB_S 328   // W^T panel stride
#define SB_S 24    // s panel stride
#define GB_S 321   // G stride
#define HB_S 321   // scores stride

__global__ __launch_bounds__(320)
void rnnattn_kernel(const float* __restrict__ s,
                    const float* __restrict__ futrue,
                    const float* __restrict__ W,
                    const float* __restrict__ bias,
                    float* __restrict__ out)
{
    __shared__ float Gbuf[16 * GB_S];              // 16x320 G strip
    __shared__ float uni[16 * FB_S + 16 * SB_S];   // overlay: (Fb+Sb) | Wb | Hb
    __shared__ float waccS[NN];
    __shared__ float csS[NN];
    __shared__ float bshS[NN];

    float* Fb = uni;                // [16][FB_S]
    float* Sb = uni + 16 * FB_S;    // [16][SB_S]
    float* Wb = uni;                // [16][WB_S]  (phase B)
    float* Hb = uni;                // [16][HB_S]  (softmax staging)

    const int tid  = threadIdx.x;
    const int wv   = tid >> 5;      // 10 waves
    const int lane = tid & 31;
    const int m    = lane & 15;
    const int ks   = lane >> 4;     // K-half select (0: K=0,1  1: K=2,3)
    const int i    = blockIdx.x;
    const float* F = futrue + (size_t)i * (NN * NN);

    // tiles this wave owns: j-tile columns jn = wv and wv+10 (em row = 0)
    const int tjn[2] = { wv, wv + 10 };

    // hoisted staging-loop decompositions (strides divide the thread count)
    const int f_r0 = tid / 80;            // F stage: row base, +4 per pass
    const int f_c4 = (tid % 80) * 4;      // F stage: column (loop-invariant)
    const int w_j0 = tid >> 4;            // W stage: j base, +20 per pass
    const int w_k  = tid & 15;            // W stage: k (loop-invariant)
    const int p_r  = tid / 20;            // prefetch: F row
    const int p_c  = (tid % 20) * 16;     // prefetch: F col (one 64B line each)

    // ---- phase 0: cs[e] = sum_t s[t,e]; preload bias; zero w accumulator
    {
        const int e = tid; // 320 threads == 320 columns
        float a = 0.f;
        for (int t = 0; t < NN; ++t) a += s[t * NN + e];
        csS[e]   = a;
        bshS[e]  = bias[e];
        waccS[e] = 0.f;
    }

    for (int strip = 0; strip < 20; ++strip) {
        const int e0 = strip * 16;

        // ================= phase A: G = s^T @ F_i  (16 x 320) =================
        v8f acc[2] = {};
        for (int t0 = 0; t0 < NN; t0 += 16) {
            __syncthreads();
            // stage F[t0:t0+16, :] (16x320) as float4 (4 rows apart per pass)
#pragma unroll
            for (int n = 0; n < 4; ++n) {
                const int r = f_r0 + 4 * n;
                const float4 v = *(const float4*)(F + (size_t)(t0 + r) * NN + f_c4);
                *(float4*)(Fb + r * FB_S + f_c4) = v;
            }
            // stage s[t0:t0+16, e0:e0+16]
            if (tid < 256) {
                const int r = tid >> 4, c = tid & 15;
                Sb[r * SB_S + c] = s[(t0 + r) * NN + e0 + c];
            }
            __syncthreads();
            // prefetch next F panel (one 64B line per thread covers 16x320 f32)
            if (t0 + 16 < NN)
                __builtin_prefetch(F + (size_t)(t0 + 16 + p_r) * NN + p_c, 0, 1);
#pragma unroll
            for (int t = 0; t < 2; ++t) {
                const int jn = tjn[t];
#pragma unroll
                for (int kk = 0; kk < 4; ++kk) {
                    const int r0 = kk * 4 + ks * 2;
                    v2f a, bv;
                    a.x  = Sb[(r0 + 0) * SB_S + m];           // A[m, k]  (A = s^T strip)
                    a.y  = Sb[(r0 + 1) * SB_S + m];
                    bv.x = Fb[(r0 + 0) * FB_S + jn * 16 + m]; // B[k, n]
                    bv.y = Fb[(r0 + 1) * FB_S + jn * 16 + m];
                    acc[t] = __builtin_amdgcn_wmma_f32_16x16x4_f32(
                        false, a, false, bv, (short)0, acc[t], false, false);
                }
            }
        }
        // spill G strip to LDS (C/D layout: row = r + 8*ks, col = lane&15)
#pragma unroll
        for (int t = 0; t < 2; ++t) {
            const int jn = tjn[t];
#pragma unroll
            for (int r = 0; r < 8; ++r)
                Gbuf[(r + 8 * ks) * GB_S + jn * 16 + m] = acc[t][r];
        }

        // ================= phase B: H = G @ W^T  (16 x 320) ==================
        v8f hcc[2] = {};
        for (int k0 = 0; k0 < NN; k0 += 16) {
            __syncthreads(); // G visible; Wb overwrite safe
            // stage W[:, k0:k0+16] transposed: Wb[k][j] = W[j*320 + k0 + k]
#pragma unroll
            for (int n = 0; n < 16; ++n) {
                const int j = w_j0 + 20 * n;
                Wb[w_k * WB_S + j] = W[j * NN + k0 + w_k];
            }
            __syncthreads();
            // prefetch next W column-panel (64B line holds k0+16..k0+31 of row tid)
            if (k0 + 16 < NN)
                __builtin_prefetch(W + (size_t)tid * NN + k0 + 16, 0, 1);
#pragma unroll
            for (int t = 0; t < 2; ++t) {
                const int jn = tjn[t];
#pragma unroll
                for (int kk = 0; kk < 4; ++kk) {
                    const int kl = kk * 4 + ks * 2;
                    v2f a, bv;
                    a.x  = Gbuf[m * GB_S + k0 + kl + 0];      // A[e=m, k=j']
                    a.y  = Gbuf[m * GB_S + k0 + kl + 1];
                    bv.x = Wb[(kl + 0) * WB_S + jn * 16 + m]; // B[k=j', n=j]
                    bv.y = Wb[(kl + 1) * WB_S + jn * 16 + m];
                    hcc[t] = __builtin_amdgcn_wmma_f32_16x16x4_f32(
                        false, a, false, bv, (short)0, hcc[t], false, false);
                }
            }
        }
        __syncthreads();
        // scores = H + cs[e] * b[j]  -> Hb
#pragma unroll
        for (int t = 0; t < 2; ++t) {
            const int jn = tjn[t];
#pragma unroll
            for (int r = 0; r < 8; ++r) {
                const int el = r + 8 * ks;
                const int j  = jn * 16 + m;
                Hb[el * HB_S + j] = hcc[t][r] + csS[e0 + el] * bshS[j];
            }
        }
        __syncthreads();
        // row softmax over j (length 320), accumulate w[j] += p[e,j]
        for (int el = wv; el < 16; el += 10) {
            float pv[10];
            float mx = -3.402823466e38f;
#pragma unroll
            for (int p = 0; p < 10; ++p) {
                pv[p] = Hb[el * HB_S + lane + 32 * p];
                mx = fmaxf(mx, pv[p]);
            }
            for (int off = 16; off >= 1; off >>= 1) mx = fmaxf(mx, __shfl_xor(mx, off));
            float sum = 0.f;
#pragma unroll
            for (int p = 0; p < 10; ++p) { pv[p] = __expf(pv[p] - mx); sum += pv[p]; }
            for (int off = 16; off >= 1; off >>= 1) sum += __shfl_xor(sum, off);
            const float inv = 1.0f / sum;
#pragma unroll
            for (int p = 0; p < 10; ++p)
                atomicAdd(&waccS[lane + 32 * p], pv[p] * inv);
        }
    }

    __syncthreads();
    // out[i, k] = sum_j w[j] * s[j, k]
    {
        const int k = tid;
        float a = 0.f;
        for (int j = 0; j < NN; ++j) a += waccS[j] * s[j * NN + k];
        out[i * NN + k] = a;
    }
}

extern "C" void kernel_launch(void* const* d_in, const int* in_sizes, int n_in,
                              void* d_out, int out_size, void* d_ws, size_t ws_size,
                              hipStream_t stream) {
    const float* s      = (const float*)d_in[0];
    const float* futrue = (const float*)d_in[1];
    const float* W      = (const float*)d_in[2];
    const float* b      = (const float*)d_in[3];
    float* out = (float*)d_out;
    rnnattn_kernel<<<dim3(320), dim3(320), 0, stream>>>(s, futrue, W, b, out);
}